// GCNModel_1391569404374
// MI455X (gfx1250) — compile-verified
//
#include <hip/hip_runtime.h>
#include <hip/hip_bf16.h>
#include <math.h>

typedef __attribute__((ext_vector_type(2))) float v2f;
typedef __attribute__((ext_vector_type(8))) float v8f;

#define IN_CH   128
#define HID_CH  64
#define OUT_CH  40
#define OUT_PAD 48
#define LDS_STRIDE (IN_CH + 4)   // pad: 132 % 64 banks = 4 -> conflict-free column reads

// ---------------------------------------------------------------- utilities
__global__ void gcn_fill_f32(float* __restrict__ p, long long n, float v) {
    long long i = (long long)blockIdx.x * blockDim.x + threadIdx.x;
    if (i < n) p[i] = v;
}

__global__ void gcn_pad_w2(const float* __restrict__ W2, float* __restrict__ Wp) {
    int idx = blockIdx.x * blockDim.x + threadIdx.x;
    if (idx >= HID_CH * OUT_PAD) return;
    int r = idx / OUT_PAD, c = idx % OUT_PAD;
    Wp[idx] = (c < OUT_CH) ? W2[r * OUT_CH + c] : 0.0f;
}

// ---------------------------------------------------------------- degree / norm
__global__ void gcn_degree(const int* __restrict__ dst, float* __restrict__ deg, long long E) {
    long long e = (long long)blockIdx.x * blockDim.x + threadIdx.x;
    if (e < E) atomicAdd(&deg[dst[e]], 1.0f);
}

__global__ void gcn_rsqrt_inplace(float* __restrict__ d, int n) {
    int i = blockIdx.x * blockDim.x + threadIdx.x;
    if (i < n) d[i] = rsqrtf(d[i]);
}

__global__ void gcn_edge_norm(const int* __restrict__ src, const int* __restrict__ dst,
                              const float* __restrict__ dis, float* __restrict__ norm,
                              long long E) {
    long long e = (long long)blockIdx.x * blockDim.x + threadIdx.x;
    if (e < E) norm[e] = dis[src[e]] * dis[dst[e]];
}

// ---------------------------------------------------------------- GEMM1: H = X @ W1  (N x 128 @ 128 x 64)
// Block = 128 threads = 4 waves. Block owns 16 rows; wave w owns cols [16w, 16w+16).
__global__ void gcn_gemm1_wmma(const float* __restrict__ X, const float* __restrict__ W,
                               float* __restrict__ H) {
    __shared__ float sA[16 * LDS_STRIDE];
    const int tid = threadIdx.x;
    const int m0  = blockIdx.x * 16;

    // cooperative load of the 16x128 A tile (coalesced float4s)
    {
        const int r = tid >> 3;          // 0..15
        const int s = (tid & 7) * 16;    // 16 floats per thread
        const float4* gp = (const float4*)(X + (size_t)(m0 + r) * IN_CH + s);
        float4* lp = (float4*)(sA + r * LDS_STRIDE + s);  // 528*r + 64*k bytes: 16B aligned
#pragma unroll
        for (int j = 0; j < 4; ++j) lp[j] = gp[j];
    }
    __syncthreads();

    const int wave  = tid >> 5;
    const int lane  = tid & 31;
    const int col   = wave * 16 + (lane & 15);
    const int khalf = (lane >> 4) << 1;             // 0 or 2 (A-matrix K pair per ISA layout)
    const float* arow = sA + (lane & 15) * LDS_STRIDE;

    v8f acc = {};
#pragma unroll
    for (int k0 = 0; k0 < IN_CH; k0 += 4) {
        v2f a, b;
        a.x = arow[k0 + khalf];
        a.y = arow[k0 + khalf + 1];
        b.x = W[(k0 + khalf)     * HID_CH + col];
        b.y = W[(k0 + khalf + 1) * HID_CH + col];
        acc = __builtin_amdgcn_wmma_f32_16x16x4_f32(false, a, false, b,
                                                    (short)0, acc, false, false);
    }

    const int mh = (lane >> 4) << 3;                // C/D: VGPR r holds M = r + 8*(lane>=16)
#pragma unroll
    for (int r = 0; r < 8; ++r)
        H[(size_t)(m0 + r + mh) * HID_CH + col] = acc[r];
}

// ---------------------------------------------------------------- GEMM2: G = H2 @ W2p  (N x 64 @ 64 x 48pad -> store 40)
// Block = 96 threads = 3 waves (3 x 16 = 48 padded cols).
__global__ void gcn_gemm2_wmma(const float* __restrict__ H, const float* __restrict__ Wp,
                               float* __restrict__ G) {
    const int tid   = threadIdx.x;
    const int wave  = tid >> 5;
    const int lane  = tid & 31;
    const int m0    = blockIdx.x * 16;
    const int col   = wave * 16 + (lane & 15);
    const int khalf = (lane >> 4) << 1;
    const float* hrow = H + (size_t)(m0 + (lane & 15)) * HID_CH;

    v8f acc = {};
#pragma unroll
    for (int k0 = 0; k0 < HID_CH; k0 += 4) {
        v2f a, b;
        a.x = hrow[k0 + khalf];
        a.y = hrow[k0 + khalf + 1];
        b.x = Wp[(k0 + khalf)     * OUT_PAD + col];
        b.y = Wp[(k0 + khalf + 1) * OUT_PAD + col];
        acc = __builtin_amdgcn_wmma_f32_16x16x4_f32(false, a, false, b,
                                                    (short)0, acc, false, false);
    }

    if (col < OUT_CH) {
        const int mh = (lane >> 4) << 3;
#pragma unroll
        for (int r = 0; r < 8; ++r)
            G[(size_t)(m0 + r + mh) * OUT_CH + col] = acc[r];
    }
}

// ---------------------------------------------------------------- edge aggregation (scatter-add)
template <int C>
__global__ void gcn_aggregate(const float* __restrict__ H, const int* __restrict__ src,
                              const int* __restrict__ dst, const float* __restrict__ norm,
                              float* __restrict__ A, long long total) {
    long long idx = (long long)blockIdx.x * blockDim.x + threadIdx.x;
    if (idx >= total) return;
    long long e = idx / C;
    int c = (int)(idx % C);
    float v = H[(size_t)src[e] * C + c] * norm[e];
    atomicAdd(&A[(size_t)dst[e] * C + c], v);
}

// ---------------------------------------------------------------- layer1 epilogue: A1 = relu(A1 + H*dis^2 + b1)
__global__ void gcn_bias_relu_self(float* __restrict__ A, const float* __restrict__ H,
                                   const float* __restrict__ dis, const float* __restrict__ b,
                                   long long total) {
    long long idx = (long long)blockIdx.x * blockDim.x + threadIdx.x;
    if (idx >= total) return;
    int i = (int)(idx / HID_CH);
    int c = (int)(idx % HID_CH);
    float d = dis[i];
    float v = A[idx] + H[idx] * d * d + b[c];
    A[idx] = fmaxf(v, 0.0f);
}

// ---------------------------------------------------------------- layer2 epilogue + log_softmax, wave per node
__global__ void gcn_logsoftmax(const float* __restrict__ A2, const float* __restrict__ G,
                               const float* __restrict__ dis, const float* __restrict__ b2,
                               float* __restrict__ out, int N) {
    const int wave = threadIdx.x >> 5;
    const int lane = threadIdx.x & 31;
    const int i = blockIdx.x * (blockDim.x >> 5) + wave;
    if (i >= N) return;

    const float d  = dis[i];
    const float d2 = d * d;
    const size_t base = (size_t)i * OUT_CH;

    float v0 = A2[base + lane] + G[base + lane] * d2 + b2[lane];
    const bool hi = (lane + 32) < OUT_CH;     // lanes 0..7 carry channels 32..39
    float v1 = hi ? (A2[base + lane + 32] + G[base + lane + 32] * d2 + b2[lane + 32])
                  : -__builtin_inff();

    float m = fmaxf(v0, v1);
#pragma unroll
    for (int off = 16; off >= 1; off >>= 1) m = fmaxf(m, __shfl_xor(m, off, 32));

    float s = expf(v0 - m) + (hi ? expf(v1 - m) : 0.0f);
#pragma unroll
    for (int off = 16; off >= 1; off >>= 1) s += __shfl_xor(s, off, 32);

    const float L = m + logf(s);
    out[base + lane] = v0 - L;
    if (hi) out[base + lane + 32] = v1 - L;
}

// ================================================================ launch
extern "C" void kernel_launch(void* const* d_in, const int* in_sizes, int n_in,
                              void* d_out, int out_size, void* d_ws, size_t ws_size,
                              hipStream_t stream) {
    const float* x  = (const float*)d_in[0];
    const float* W1 = (const float*)d_in[1];
    const float* b1 = (const float*)d_in[2];
    const float* W2 = (const float*)d_in[3];
    const float* b2 = (const float*)d_in[4];
    const int*   ei = (const int*)d_in[5];    // harness contract: integer -> const int*

    const int       N = in_sizes[0] / IN_CH;      // 100000 (divisible by 16)
    const long long E = in_sizes[5] / 2;          // 1600000
    const int* src = ei;
    const int* dst = ei + E;

    // workspace layout (floats)
    float* ws   = (float*)d_ws;
    float* deg  = ws;                               // N   (deg -> dis in place)
    float* norm = deg  + N;                         // E
    float* H1   = norm + E;                         // N*HID_CH
    float* A1   = H1   + (size_t)N * HID_CH;        // N*HID_CH (-> relu'd hidden)
    float* G    = A1   + (size_t)N * HID_CH;        // N*OUT_CH
    float* A2   = G    + (size_t)N * OUT_CH;        // N*OUT_CH
    float* Wp   = A2   + (size_t)N * OUT_CH;        // HID_CH*OUT_PAD
    float* out  = (float*)d_out;

    const int T = 256;
    auto blk = [](long long n, int t) { return (unsigned)((n + t - 1) / t); };

    // init (every call: deterministic, no cross-call state)
    gcn_fill_f32<<<blk(N, T), T, 0, stream>>>(deg, N, 1.0f);                 // self-loop degree
    gcn_fill_f32<<<blk((long long)N * HID_CH, T), T, 0, stream>>>(A1, (long long)N * HID_CH, 0.0f);
    gcn_fill_f32<<<blk((long long)N * OUT_CH, T), T, 0, stream>>>(A2, (long long)N * OUT_CH, 0.0f);
    gcn_pad_w2<<<blk(HID_CH * OUT_PAD, T), T, 0, stream>>>(W2, Wp);

    // degree -> dis -> per-edge norm
    gcn_degree<<<blk(E, T), T, 0, stream>>>(dst, deg, E);
    gcn_rsqrt_inplace<<<blk(N, T), T, 0, stream>>>(deg, N);
    gcn_edge_norm<<<blk(E, T), T, 0, stream>>>(src, dst, deg, norm, E);

    // layer 1: H1 = x @ W1 ; A1 += scatter(H1*norm) ; A1 = relu(A1 + self + b1)
    gcn_gemm1_wmma<<<N / 16, 128, 0, stream>>>(x, W1, H1);
    gcn_aggregate<HID_CH><<<blk(E * HID_CH, T), T, 0, stream>>>(H1, src, dst, norm, A1, E * HID_CH);
    gcn_bias_relu_self<<<blk((long long)N * HID_CH, T), T, 0, stream>>>(A1, H1, deg, b1,
                                                                        (long long)N * HID_CH);

    // layer 2: G = A1 @ W2 ; A2 += scatter(G*norm) ; out = log_softmax(A2 + self + b2)
    gcn_gemm2_wmma<<<N / 16, 96, 0, stream>>>(A1, Wp, G);
    gcn_aggregate<OUT_CH><<<blk(E * OUT_CH, T), T, 0, stream>>>(G, src, dst, norm, A2, E * OUT_CH);
    gcn_logsoftmax<<<blk(N, 8 * 32) /* 8 waves per block */, 256, 0, stream>>>(A2, G, deg, b2, out, N);
}